// Encoder_7894149890238
// MI455X (gfx1250) — compile-verified
//
#include <hip/hip_runtime.h>
#include <math.h>

typedef float v2f __attribute__((ext_vector_type(2)));
typedef float v8f __attribute__((ext_vector_type(8)));

#define BATCH 192
#define DIN   2048
#define HID   128

__device__ __forceinline__ float relu_(float x) { return x > 0.f ? x : 0.f; }

// ---------------------------------------------------------------------------
// Generic 16x16-tile GEMM:  O[m,n] = relu( sum_k A[m,k]*W[n,k] + bias[n] )
// A: [M,K] row-major, W: [N,K] row-major (i.e. torch Linear weight layout).
// One wave (32 threads) computes one 16x16 output tile using
// V_WMMA_F32_16X16X4_F32, accumulating over K in steps of 4.
//
// A-fragment layout (ISA 7.12.2, 32-bit A 16x4): lane L holds
//   M = L%16, K = (L/16)*2 + v  for VGPR v in {0,1}  -> one b64 load per step.
// B-fragment (4x16) mirrors it with N = L%16 striped across lanes.
// C/D layout: lane L, VGPR r -> M = (L/16)*8 + r, N = L%16.
// ---------------------------------------------------------------------------
template <int K, int N>
__global__ __launch_bounds__(32) void k_gemm(
    const float* __restrict__ A0, const float* __restrict__ A1,
    const float* __restrict__ W0, const float* __restrict__ Bi0,
    const float* __restrict__ W1, const float* __restrict__ Bi1,
    float* __restrict__ O0, float* __restrict__ O1) {
  const int lane = threadIdx.x;
  const int tm = blockIdx.x, tn = blockIdx.y, head = blockIdx.z;

  const float* A    = head ? A1  : A0;
  const float* W    = head ? W1  : W0;
  const float* bias = head ? Bi1 : Bi0;
  float*       O    = head ? O1  : O0;

  const int koff = (lane >> 4) * 2;
  const float* ap = A + (size_t)(tm * 16 + (lane & 15)) * K + koff;
  const float* wp = W + (size_t)(tn * 16 + (lane & 15)) * K + koff;

  v8f acc = {};
#pragma unroll 4
  for (int k = 0; k < K; k += 4) {
    v2f a = *(const v2f*)(ap + k);
    v2f b = *(const v2f*)(wp + k);
    acc = __builtin_amdgcn_wmma_f32_16x16x4_f32(
        /*neg_a=*/false, a, /*neg_b=*/false, b,
        /*c_mod=*/(short)0, acc, /*reuse_a=*/false, /*reuse_b=*/false);
  }

  const int n     = tn * 16 + (lane & 15);
  const int mbase = tm * 16 + ((lane >> 4) << 3);
  const float bv  = bias[n];
#pragma unroll
  for (int r = 0; r < 8; ++r) {
    O[(size_t)(mbase + r) * N + n] = relu_(acc[r] + bv);
  }
}

// ---------------------------------------------------------------------------
// Block-wide sum reduction (256 threads, wave32). Result valid in thread 0.
// ---------------------------------------------------------------------------
__device__ __forceinline__ float block_reduce_sum_256(float v) {
  __shared__ float sm[8];
#pragma unroll
  for (int off = 16; off > 0; off >>= 1) v += __shfl_xor(v, off, 32);
  const int wid = threadIdx.x >> 5;
  if ((threadIdx.x & 31) == 0) sm[wid] = v;
  __syncthreads();
  if (threadIdx.x < 8) {
    v = sm[threadIdx.x];
#pragma unroll
    for (int off = 4; off > 0; off >>= 1) v += __shfl_xor(v, off, 32);
  }
  return v;
}

// ---------------------------------------------------------------------------
// From lv[j,d]: var = exp(0.5*lv); store scale = sqrt(var), inv_var = 1/var,
// and lvsum[j] = sum_d lv[j,d].  One block per row j.
// ---------------------------------------------------------------------------
__global__ __launch_bounds__(256) void k_lvpost(
    const float* __restrict__ lv, float* __restrict__ scale,
    float* __restrict__ inv_var, float* __restrict__ lvsum) {
  const int j = blockIdx.x;
  const int t = threadIdx.x;
  const size_t base = (size_t)j * DIN;
  float s = 0.f;
#pragma unroll
  for (int c = 0; c < DIN / 256; ++c) {
    const int d   = t + c * 256;
    const float l = lv[base + d];
    const float var = expf(0.5f * l);
    scale[base + d]   = sqrtf(var);
    inv_var[base + d] = 1.0f / var;
    s += l;
  }
  s = block_reduce_sum_256(s);
  if (t == 0) lvsum[j] = s;
}

// ---------------------------------------------------------------------------
// Main streaming pass: one block per (i,j) pair.
//   p[i,j,d]  = mu[j,d] + eps[i,j,d]*scale[j,d]
//   maha[i,j] = sum_d (p-mu)^2 * inv_var
//   logp[i,j] = -0.5*(maha + D*log(2pi)) - 0.25*sum_d lv[j,d]
// Pure bandwidth: read eps once (302 MB), write p once (302 MB); per-j vectors
// (mu/scale/inv_var, 1.5 MB each) stay resident in the 192 MB L2.
// ---------------------------------------------------------------------------
__global__ __launch_bounds__(256) void k_main(
    const float* __restrict__ eps, const float* __restrict__ mu,
    const float* __restrict__ scale, const float* __restrict__ inv_var,
    const float* __restrict__ lvsum, float* __restrict__ p_out,
    float* __restrict__ logp_out) {
  const int pair = blockIdx.x;          // i*BATCH + j
  const int j    = pair % BATCH;
  const size_t base  = (size_t)pair * DIN;
  const size_t jbase = (size_t)j * DIN;
  const int t = threadIdx.x;

  const float4* e4  = (const float4*)(eps + base);
  const float4* m4  = (const float4*)(mu + jbase);
  const float4* s4  = (const float4*)(scale + jbase);
  const float4* iv4 = (const float4*)(inv_var + jbase);
  float4*       p4  = (float4*)(p_out + base);

  float acc = 0.f;
#pragma unroll
  for (int c = 0; c < (DIN / 4) / 256; ++c) {
    const int idx = t + c * 256;
    const float4 e  = e4[idx];
    const float4 m  = m4[idx];
    const float4 s  = s4[idx];
    const float4 iv = iv4[idx];
    float4 p;
    p.x = fmaf(e.x, s.x, m.x);
    p.y = fmaf(e.y, s.y, m.y);
    p.z = fmaf(e.z, s.z, m.z);
    p.w = fmaf(e.w, s.w, m.w);
    p4[idx] = p;
    const float dx = p.x - m.x, dy = p.y - m.y, dz = p.z - m.z, dw = p.w - m.w;
    acc += dx * dx * iv.x + dy * dy * iv.y + dz * dz * iv.z + dw * dw * iv.w;
  }

  acc = block_reduce_sum_256(acc);
  if (t == 0) {
    const float d_log2pi = 2048.0f * 1.8378770664093453f;
    logp_out[pair] = -0.5f * (acc + d_log2pi) - 0.25f * lvsum[j];
  }
}

// ---------------------------------------------------------------------------
extern "C" void kernel_launch(void* const* d_in, const int* in_sizes, int n_in,
                              void* d_out, int out_size, void* d_ws,
                              size_t ws_size, hipStream_t stream) {
  const float* q     = (const float*)d_in[0];
  const float* eps   = (const float*)d_in[1];
  const float* mu_w1 = (const float*)d_in[2];
  const float* mu_b1 = (const float*)d_in[3];
  const float* mu_w2 = (const float*)d_in[4];
  const float* mu_b2 = (const float*)d_in[5];
  const float* mu_w3 = (const float*)d_in[6];
  const float* mu_b3 = (const float*)d_in[7];
  const float* lv_w1 = (const float*)d_in[8];
  const float* lv_b1 = (const float*)d_in[9];
  const float* lv_w2 = (const float*)d_in[10];
  const float* lv_b2 = (const float*)d_in[11];
  const float* lv_w3 = (const float*)d_in[12];
  const float* lv_b3 = (const float*)d_in[13];

  // Workspace layout (floats). Total ~6.4 MB.
  float* ws      = (float*)d_ws;
  float* h1      = ws;                          // [2][192][128]
  float* h2      = h1 + 2 * BATCH * HID;        // [2][192][128]
  float* lv      = h2 + 2 * BATCH * HID;        // [192][2048]
  float* mu      = lv + BATCH * DIN;            // [192][2048]
  float* scale   = mu + BATCH * DIN;            // [192][2048]
  float* inv_var = scale + BATCH * DIN;         // [192][2048]
  float* lvsum   = inv_var + BATCH * DIN;       // [192]

  float* p_out = (float*)d_out;                       // [192][192][2048]
  float* logp  = p_out + (size_t)BATCH * BATCH * DIN; // [192][192]

  // Layer 1: [192,2048] x [128,2048]^T -> [192,128], both heads (z).
  k_gemm<DIN, HID><<<dim3(BATCH / 16, HID / 16, 2), 32, 0, stream>>>(
      q, q, mu_w1, mu_b1, lv_w1, lv_b1, h1, h1 + BATCH * HID);
  // Layer 2: [192,128] x [128,128]^T -> [192,128].
  k_gemm<HID, HID><<<dim3(BATCH / 16, HID / 16, 2), 32, 0, stream>>>(
      h1, h1 + BATCH * HID, mu_w2, mu_b2, lv_w2, lv_b2, h2, h2 + BATCH * HID);
  // Layer 3: [192,128] x [2048,128]^T -> [192,2048] (mu head -> mu, lv head -> lv).
  k_gemm<HID, DIN><<<dim3(BATCH / 16, DIN / 16, 2), 32, 0, stream>>>(
      h2, h2 + BATCH * HID, mu_w3, mu_b3, lv_w3, lv_b3, mu, lv);
  // scale / inv_var / row-sums of lv.
  k_lvpost<<<BATCH, 256, 0, stream>>>(lv, scale, inv_var, lvsum);
  // Streaming p + log_prob.
  k_main<<<BATCH * BATCH, 256, 0, stream>>>(eps, mu, scale, inv_var, lvsum,
                                            p_out, logp);
}